// Pnet_49460843381002
// MI455X (gfx1250) — compile-verified
//
#include <hip/hip_runtime.h>
#include <hip/hip_bf16.h>
#include <math.h>

typedef float v2f __attribute__((ext_vector_type(2)));
typedef float v8f __attribute__((ext_vector_type(8)));

#if defined(__has_builtin)
#if __has_builtin(__builtin_amdgcn_wmma_f32_16x16x4_f32)
#define HAVE_WMMA_F32_16X16X4 1
#endif
#endif

// ---------------- problem constants ----------------
#define IN_W   2048
#define H1_W   1023         // after conv1(3x3)+pool2
#define H2_W   1021         // after conv2(3x3)
#define H3_W   1019         // after conv3(3x3)
#define NP3    (H3_W * H3_W)
#define TOPK   512
#define NBUCK  2048
#define CANDCAP 4096
#define THRESH 0.6f

// =====================================================================
// K1: conv1 (3->10, 3x3, VALID) + PReLU + maxpool 2x2  -> h1[10][1023][1023]
// =====================================================================
__global__ __launch_bounds__(256) void conv1_prelu_pool(
    const float* __restrict__ x, const float* __restrict__ w1,
    const float* __restrict__ b1, const float* __restrict__ p1,
    float* __restrict__ h1)
{
    int px = blockIdx.x * blockDim.x + threadIdx.x;
    int py = blockIdx.y;
    if (px >= H1_W) return;
    const int HW = IN_W * IN_W;

    float patch[3][4][4];
    const float* xp = x + (2 * py) * IN_W + 2 * px;
#pragma unroll
    for (int c = 0; c < 3; ++c)
#pragma unroll
        for (int r = 0; r < 4; ++r)
#pragma unroll
            for (int q = 0; q < 4; ++q)
                patch[c][r][q] = xp[c * HW + r * IN_W + q];

#pragma unroll
    for (int co = 0; co < 10; ++co) {
        float a = p1[co];
        float mx = -INFINITY;
#pragma unroll
        for (int dy = 0; dy < 2; ++dy)
#pragma unroll
            for (int dx = 0; dx < 2; ++dx) {
                float acc = b1[co];
#pragma unroll
                for (int c = 0; c < 3; ++c)
#pragma unroll
                    for (int ky = 0; ky < 3; ++ky)
#pragma unroll
                        for (int kx = 0; kx < 3; ++kx)
                            acc = fmaf(patch[c][dy + ky][dx + kx],
                                       w1[((co * 3 + c) * 3 + ky) * 3 + kx], acc);
                float v = (acc >= 0.f) ? acc : a * acc;
                mx = fmaxf(mx, v);
            }
        h1[co * (H1_W * H1_W) + py * H1_W + px] = mx;
    }
}

// =====================================================================
// K2/K3: 3x3 conv + PReLU as implicit GEMM on V_WMMA_F32_16X16X4_F32.
//   Padded-K scheme: logical K' = CIN*12, one (cin,ky) pair per K-group of 4,
//   kx in {0,1,2,pad}.  B rows with kx==3 are zeroed in LDS, so the A element
//   hi-lanes fetch for the pad slot is annihilated without any masking.
//   A-operand addresses are a single per-lane pointer (ap2 = row + m + ksel)
//   plus compile-time immediate offsets -> no per-step address VALU.
// =====================================================================
template <int CIN, int COUT, int WIN>
__global__ __launch_bounds__(256) void conv3x3_kernel(
    const float* __restrict__ in, float* __restrict__ out,
    const float* __restrict__ w, const float* __restrict__ bias,
    const float* __restrict__ pr)
{
    constexpr int WOUT = WIN - 2;
    constexpr int KC   = CIN * 9;       // dense K
    constexpr int KP   = CIN * 12;      // padded K (4 per (cin,ky))
    constexpr int HWIN = WIN * WIN;
    constexpr int NPO  = WOUT * WOUT;
    constexpr int NT   = COUT / 16;

    __shared__ float Blds[KP * COUT];
    int tid = threadIdx.x;
    for (int i = tid; i < KP * COUT; i += 256) {
        int k = i / COUT, n = i % COUT;
        int cin = k / 12, rem = k % 12;
        int ky = rem >> 2, kx = rem & 3;
        Blds[i] = (kx < 3) ? w[n * KC + cin * 9 + ky * 3 + kx] : 0.f;
    }
    __syncthreads();

    int wave  = tid >> 5;
    int lane  = tid & 31;
    int xtile = blockIdx.x * 8 + wave;
    int y     = blockIdx.y;
    int x0    = xtile * 16;
    if (x0 > WOUT - 16) x0 = WOUT - 16;     // clamp last tile (overlap recompute)

    int m    = lane & 15;
    int ksel = (lane >> 4) * 2;             // 0 for lanes 0-15, 2 for lanes 16-31
    int ncol = lane & 15;
    const float* ap2 = in + y * WIN + x0 + m + ksel;   // per-lane A base
    int bbase = ksel * COUT + ncol;                    // per-lane B base (elems)

#if HAVE_WMMA_F32_16X16X4
    v8f acc[NT];
#pragma unroll
    for (int t = 0; t < NT; ++t) {
        float bv = bias[ncol + t * 16];
#pragma unroll
        for (int r = 0; r < 8; ++r) acc[t][r] = bv;
    }

    const float* ap_c = ap2;
#pragma unroll
    for (int cin = 0; cin < CIN; ++cin) {
#pragma unroll
        for (int ky = 0; ky < 3; ++ky) {
            const int s = cin * 3 + ky;       // compile-time after unroll
            v2f a;
            a.x = ap_c[ky * WIN];             // kx = ksel   (imm offset)
            a.y = ap_c[ky * WIN + 1];         // kx = ksel+1 (pad slot for hi lanes)
#pragma unroll
            for (int t = 0; t < NT; ++t) {
                v2f b;
                b.x = Blds[bbase + (4 * s)     * COUT + t * 16];
                b.y = Blds[bbase + (4 * s + 1) * COUT + t * 16];
                acc[t] = __builtin_amdgcn_wmma_f32_16x16x4_f32(
                    false, a, false, b, (short)0, acc[t], false, false);
            }
        }
        ap_c += HWIN;
    }

    // D layout: VGPR r <-> M = r (lanes 0-15) / r+8 (lanes 16-31), N = lane%16.
    int mbase = (lane >> 4) * 8;
#pragma unroll
    for (int t = 0; t < NT; ++t) {
        int n    = ncol + t * 16;
        float pv = pr[n];
        float* op = out + n * NPO + y * WOUT + x0 + mbase;
#pragma unroll
        for (int r = 0; r < 8; ++r) {
            float v = acc[t][r];
            op[r] = (v >= 0.f) ? v : pv * v;
        }
    }
#else
    // scalar fallback (kept only so the file always compiles)
    int half = lane >> 4;
    for (int n = half * (COUT / 2); n < (half + 1) * (COUT / 2); ++n) {
        float accv = bias[n];
        for (int cin = 0; cin < CIN; ++cin)
            for (int ky = 0; ky < 3; ++ky)
                for (int kx = 0; kx < 3; ++kx)
                    accv = fmaf(in[cin * HWIN + (y + ky) * WIN + x0 + m + kx],
                                Blds[(cin * 12 + ky * 4 + kx) * COUT + n], accv);
        float pv = pr[n];
        out[n * NPO + y * WOUT + x0 + m] = (accv >= 0.f) ? accv : pv * accv;
    }
#endif
}

// =====================================================================
// K0: zero histogram + meta every call (replay-safe)
// =====================================================================
__global__ void zero_scratch(int* hist, int* meta)
{
    int i = blockIdx.x * blockDim.x + threadIdx.x;
    if (i < NBUCK) hist[i] = 0;
    if (i < 16)    meta[i] = 0;
}

// =====================================================================
// K4: 1x1 heads + softmax(ch1) + histogram of scores >= 0.6
// =====================================================================
__global__ __launch_bounds__(256) void heads_kernel(
    const float* __restrict__ h3,
    const float* __restrict__ wa, const float* __restrict__ ba,
    const float* __restrict__ wb, const float* __restrict__ bb,
    float* __restrict__ scores, float* __restrict__ reg4, int* __restrict__ hist)
{
    int p = blockIdx.x * blockDim.x + threadIdx.x;
    if (p >= NP3) return;
    float c0 = ba[0], c1 = ba[1];
    float r0 = bb[0], r1 = bb[1], r2 = bb[2], r3 = bb[3];
#pragma unroll
    for (int c = 0; c < 32; ++c) {
        float v = h3[c * NP3 + p];
        c0 = fmaf(v, wa[c], c0);
        c1 = fmaf(v, wa[32 + c], c1);
        r0 = fmaf(v, wb[c], r0);
        r1 = fmaf(v, wb[32 + c], r1);
        r2 = fmaf(v, wb[64 + c], r2);
        r3 = fmaf(v, wb[96 + c], r3);
    }
    float s = 1.f / (1.f + expf(c0 - c1));   // softmax channel-1
    scores[p] = s;
    reg4[4 * p + 0] = r0;
    reg4[4 * p + 1] = r1;
    reg4[4 * p + 2] = r2;
    reg4[4 * p + 3] = r3;
    if (s >= THRESH) {
        int b = (int)((s - THRESH) * 5120.f);   // 2048 buckets over [0.6, 1.0]
        b = b < 0 ? 0 : (b > NBUCK - 1 ? NBUCK - 1 : b);
        atomicAdd(&hist[b], 1);
    }
}

// K5: find bucket cutoff such that >= TOPK candidates survive
__global__ void cutoff_kernel(const int* __restrict__ hist, int* meta)
{
    if (threadIdx.x == 0) {
        int total = 0, cb = 0;
        for (int b = NBUCK - 1; b >= 0; --b) {
            total += hist[b];
            if (total >= TOPK) { cb = b; break; }
        }
        ((float*)meta)[1] = THRESH + cb * (1.f / 5120.f);
    }
}

// K6: compact candidates above cutoff into packed (score, ~idx) keys
__global__ __launch_bounds__(256) void compact_kernel(
    const float* __restrict__ scores, int* meta, unsigned long long* __restrict__ cands)
{
    int p = blockIdx.x * blockDim.x + threadIdx.x;
    if (p >= NP3) return;
    float s = scores[p];
    float thr = ((float*)meta)[1];
    if (s >= thr && s >= THRESH) {
        int slot = atomicAdd(&meta[0], 1);
        if (slot < CANDCAP) {
            unsigned long long key =
                ((unsigned long long)__float_as_uint(s) << 32) |
                (unsigned long long)(0xFFFFFFFFu - (unsigned)p);
            cands[slot] = key;
        }
    }
}

// =====================================================================
// K7: single block — bitonic sort (desc) 4096 keys, take top-512, decode
//     boxes, run both sequential NMS passes, emit 512x5 output.
// =====================================================================
__global__ __launch_bounds__(1024) void finalize_kernel(
    const unsigned long long* __restrict__ cands, const int* __restrict__ meta,
    const float* __restrict__ reg4, float* __restrict__ out)
{
    __shared__ unsigned long long skey[CANDCAP];
    __shared__ float bx1[TOPK], by1[TOPK], bx2[TOPK], by2[TOPK], ssc[TOPK], sar[TOPK];
    __shared__ int   skeep[TOPK];

    int tid = threadIdx.x;
    int nc = meta[0];
    if (nc > CANDCAP) nc = CANDCAP;

    for (int i = tid; i < CANDCAP; i += 1024)
        skey[i] = (i < nc) ? cands[i] : 0ULL;
    __syncthreads();

    // bitonic sort, descending (ties: larger ~idx first => smaller idx first)
    for (int k = 2; k <= CANDCAP; k <<= 1) {
        for (int j = k >> 1; j > 0; j >>= 1) {
            for (int i = tid; i < CANDCAP; i += 1024) {
                int ixj = i ^ j;
                if (ixj > i) {
                    unsigned long long a = skey[i], b = skey[ixj];
                    bool desc = ((i & k) == 0);
                    if (desc ? (a < b) : (a > b)) { skey[i] = b; skey[ixj] = a; }
                }
            }
            __syncthreads();
        }
    }

    if (tid < TOPK) {
        unsigned long long key = skey[tid];
        float s = __uint_as_float((unsigned)(key >> 32));
        unsigned idx = 0xFFFFFFFFu - (unsigned)(key & 0xFFFFFFFFu);
        bool valid = (key != 0ULL) && (s >= THRESH);
        float X1 = 0.f, Y1 = 0.f, X2 = 0.f, Y2 = 0.f, S = 0.f;
        if (valid) {
            float yy = (float)(idx / (unsigned)H3_W);
            float xx = (float)(idx % (unsigned)H3_W);
            // off_x = off_y = 6, stride 2, scale 0.6, win 12 -> ww = 20
            float cx = (xx * 2.f + 6.f) * (1.f / 0.6f);
            float cy = (yy * 2.f + 6.f) * (1.f / 0.6f);
            float r0 = reg4[4 * idx + 0], r1 = reg4[4 * idx + 1];
            float r2 = reg4[4 * idx + 2], r3 = reg4[4 * idx + 3];
            X1 = cx - 10.f + r0 * 20.f;
            Y1 = cy - 10.f + r1 * 20.f;
            X2 = cx + 10.f + r2 * 20.f;
            Y2 = cy + 10.f + r3 * 20.f;
            S = s;
        }
        bx1[tid] = X1; by1[tid] = Y1; bx2[tid] = X2; by2[tid] = Y2;
        ssc[tid] = S;
        sar[tid] = (X2 - X1) * (Y2 - Y1);
        skeep[tid] = valid ? 1 : 0;
    }
    __syncthreads();

    // two sequential NMS passes: thr = 0.5 then 0.7 (matches reference order)
    for (int pass = 0; pass < 2; ++pass) {
        float thr = (pass == 0) ? 0.5f : 0.7f;
        for (int i = 0; i < TOPK; ++i) {
            if (tid < TOPK && tid > i && skeep[i] && skeep[tid]) {
                float xi1 = fmaxf(bx1[i], bx1[tid]);
                float yi1 = fmaxf(by1[i], by1[tid]);
                float xi2 = fminf(bx2[i], bx2[tid]);
                float yi2 = fminf(by2[i], by2[tid]);
                float inter = fmaxf(xi2 - xi1, 0.f) * fmaxf(yi2 - yi1, 0.f);
                float iou = inter / (sar[i] + sar[tid] - inter + 1e-9f);
                if (iou > thr) skeep[tid] = 0;
            }
            __syncthreads();
        }
    }

    if (tid < TOPK) {
        float kf = skeep[tid] ? 1.f : 0.f;
        out[tid * 5 + 0] = bx1[tid] * kf;
        out[tid * 5 + 1] = by1[tid] * kf;
        out[tid * 5 + 2] = bx2[tid] * kf;
        out[tid * 5 + 3] = by2[tid] * kf;
        out[tid * 5 + 4] = ssc[tid] * kf;
    }
}

// =====================================================================
extern "C" void kernel_launch(void* const* d_in, const int* in_sizes, int n_in,
                              void* d_out, int out_size, void* d_ws, size_t ws_size,
                              hipStream_t stream)
{
    (void)in_sizes; (void)n_in; (void)out_size; (void)ws_size;
    const float* x  = (const float*)d_in[0];
    const float* w1 = (const float*)d_in[1];
    const float* b1 = (const float*)d_in[2];
    const float* p1 = (const float*)d_in[3];
    const float* w2 = (const float*)d_in[4];
    const float* b2 = (const float*)d_in[5];
    const float* p2 = (const float*)d_in[6];
    const float* w3 = (const float*)d_in[7];
    const float* b3 = (const float*)d_in[8];
    const float* p3 = (const float*)d_in[9];
    const float* wa = (const float*)d_in[10];
    const float* ba = (const float*)d_in[11];
    const float* wb = (const float*)d_in[12];
    const float* bb = (const float*)d_in[13];
    float* out = (float*)d_out;

    // workspace carve-up (floats, 64-element aligned)
    float* ws = (float*)d_ws;
    size_t off = 0;
    auto alloc = [&](size_t n) {
        float* p = ws + off;
        off += (n + 63) & ~(size_t)63;
        return p;
    };
    float* h1     = alloc((size_t)10 * H1_W * H1_W);
    float* h2     = alloc((size_t)16 * H2_W * H2_W);
    float* h3     = alloc((size_t)32 * H3_W * H3_W);
    float* scores = alloc((size_t)NP3);
    float* reg4   = alloc((size_t)4 * NP3);
    int*   hist   = (int*)alloc(NBUCK);
    int*   meta   = (int*)alloc(64);
    unsigned long long* cands = (unsigned long long*)alloc(2 * CANDCAP + 16);

    // 1) conv1 + prelu + pool
    conv1_prelu_pool<<<dim3((H1_W + 255) / 256, H1_W), 256, 0, stream>>>(x, w1, b1, p1, h1);
    // 0) zero histogram + meta
    zero_scratch<<<(NBUCK + 255) / 256, 256, 0, stream>>>(hist, meta);
    // 2) conv2 (WMMA implicit GEMM, 10->16)
    conv3x3_kernel<10, 16, H1_W><<<dim3(8, H2_W), 256, 0, stream>>>(h1, h2, w2, b2, p2);
    // 3) conv3 (WMMA implicit GEMM, 16->32)
    conv3x3_kernel<16, 32, H2_W><<<dim3(8, H3_W), 256, 0, stream>>>(h2, h3, w3, b3, p3);
    // 4) heads + softmax + histogram
    heads_kernel<<<(NP3 + 255) / 256, 256, 0, stream>>>(h3, wa, ba, wb, bb, scores, reg4, hist);
    // 5) radix-style cutoff for top-512
    cutoff_kernel<<<1, 32, 0, stream>>>(hist, meta);
    // 6) compact candidates
    compact_kernel<<<(NP3 + 255) / 256, 256, 0, stream>>>(scores, meta, cands);
    // 7) sort + boxes + NMS x2 + output
    finalize_kernel<<<1, 1024, 0, stream>>>(cands, meta, reg4, out);
}